// RankingLoss_82016695484486
// MI455X (gfx1250) — compile-verified
//
#include <hip/hip_runtime.h>

typedef float v4f __attribute__((ext_vector_type(4)));
typedef float v2f __attribute__((ext_vector_type(2)));
typedef float v8f __attribute__((ext_vector_type(8)));

#define MARGIN 0.5f
#define BLOCK 256

// ---------------------------------------------------------------------------
// Kernel 1: one block per row. Streaming top-2 (value,index) + gold gather,
// hinge loss -> d_ws[row].
// ---------------------------------------------------------------------------
__global__ __launch_bounds__(BLOCK) void rankloss_row_kernel(
    const float* __restrict__ scores, const int* __restrict__ gold,
    float* __restrict__ row_loss, int V) {
  const int row = blockIdx.x;
  const int tid = threadIdx.x;
  const float* rp = scores + (size_t)row * (size_t)V;

  float v1 = -INFINITY, v2 = -INFINITY;
  int i1 = 0x7fffffff, i2 = 0x7fffffff;

  // vectorized main body: b128 non-temporal streaming loads
  const int nvec = V >> 2;  // 12564 for V=50257
  const v4f* rp4 = (const v4f*)rp;
  for (int q = tid; q < nvec; q += BLOCK) {
    v4f x = __builtin_nontemporal_load(rp4 + q);
    int base = q << 2;
#pragma unroll
    for (int j = 0; j < 4; ++j) {
      float v = x[j];
      int idx = base + j;
      if (v > v1) {
        v2 = v1; i2 = i1;
        v1 = v;  i1 = idx;
      } else if (v > v2) {
        v2 = v;  i2 = idx;
      }
    }
  }
  // scalar tail (V % 4 elements)
  for (int idx = (nvec << 2) + tid; idx < V; idx += BLOCK) {
    float v = __builtin_nontemporal_load(rp + idx);
    if (v > v1) {
      v2 = v1; i2 = i1;
      v1 = v;  i1 = idx;
    } else if (v > v2) {
      v2 = v;  i2 = idx;
    }
  }

  // block-wide top-2 merge in LDS
  __shared__ float s_v1[BLOCK], s_v2[BLOCK];
  __shared__ int   s_i1[BLOCK], s_i2[BLOCK];
  s_v1[tid] = v1; s_v2[tid] = v2;
  s_i1[tid] = i1; s_i2[tid] = i2;
  __syncthreads();

  for (int s = BLOCK >> 1; s > 0; s >>= 1) {
    if (tid < s) {
      float a1 = s_v1[tid],     a2 = s_v2[tid];
      int   ja1 = s_i1[tid],    ja2 = s_i2[tid];
      float b1 = s_v1[tid + s], b2 = s_v2[tid + s];
      int   jb1 = s_i1[tid + s], jb2 = s_i2[tid + s];
      float m1, m2; int k1, k2;
      // prefer lower index on exact ties (argmax-first semantics)
      bool aWins = (a1 > b1) || (a1 == b1 && ja1 < jb1);
      if (aWins) {
        m1 = a1; k1 = ja1;
        if ((b1 > a2) || (b1 == a2 && jb1 < ja2)) { m2 = b1; k2 = jb1; }
        else                                      { m2 = a2; k2 = ja2; }
      } else {
        m1 = b1; k1 = jb1;
        if ((a1 > b2) || (a1 == b2 && ja1 < jb2)) { m2 = a1; k2 = ja1; }
        else                                      { m2 = b2; k2 = jb2; }
      }
      s_v1[tid] = m1; s_i1[tid] = k1;
      s_v2[tid] = m2; s_i2[tid] = k2;
    }
    __syncthreads();
  }

  if (tid == 0) {
    float max1 = s_v1[0], max2 = s_v2[0];
    int best = s_i1[0], second = s_i2[0];
    int g = gold[row];
    float gs = rp[g];
    bool sw = (best == g);
    float neg = sw ? max2 : max1;
    int sample = sw ? second : best;
    float loss = fmaxf(MARGIN + neg - gs, 0.0f);
    if (sample == g) loss = 0.0f;
    row_loss[row] = loss;
  }
}

// ---------------------------------------------------------------------------
// Kernel 2: single wave32. Sum the N row losses with an accumulating chain of
// V_WMMA_F32_16X16X4_F32 (A = ones): D[m][n] = sum_k B[k][n] + C[m][n], so
// the full-tile sum equals 16x the data sum independent of B's VGPR layout.
// ---------------------------------------------------------------------------
__global__ __launch_bounds__(32) void rankloss_finalize_kernel(
    const float* __restrict__ row_loss, float* __restrict__ out, int N) {
  const int lane = threadIdx.x;  // wave32: 0..31, EXEC all ones (WMMA req.)

  v2f a; a.x = 1.0f; a.y = 1.0f;   // A = ones 16x4 tile
  v8f acc = {};                     // C/D f32 16x16 accumulator

  const int nit = N >> 6;  // 64 values per WMMA step
  for (int it = 0; it < nit; ++it) {
    v2f b;
    b.x = row_loss[(it << 6) + lane];
    b.y = row_loss[(it << 6) + 32 + lane];
    acc = __builtin_amdgcn_wmma_f32_16x16x4_f32(
        /*neg_a=*/false, a, /*neg_b=*/false, b,
        /*c_mod=*/(short)0, acc, /*reuse_a=*/false, /*reuse_b=*/false);
  }

  float lane_sum = acc[0] + acc[1] + acc[2] + acc[3] +
                   acc[4] + acc[5] + acc[6] + acc[7];

  __shared__ float red[32];
  red[lane] = lane_sum;
  __syncthreads();

  if (lane == 0) {
    float t = 0.0f;
#pragma unroll
    for (int i = 0; i < 32; ++i) t += red[i];
    t *= (1.0f / 16.0f);  // tile sum counts each B element 16x
    // scalar tail if N not a multiple of 64 (not hit for N=4096)
    for (int idx = nit << 6; idx < N; ++idx) t += row_loss[idx];
    out[0] = t / (float)N;  // size_average
  }
}

// ---------------------------------------------------------------------------
extern "C" void kernel_launch(void* const* d_in, const int* in_sizes, int n_in,
                              void* d_out, int out_size, void* d_ws, size_t ws_size,
                              hipStream_t stream) {
  const float* scores = (const float*)d_in[0];
  const int*   gold   = (const int*)d_in[1];
  float* out = (float*)d_out;
  float* ws  = (float*)d_ws;  // B floats of per-row loss

  const int Bn = in_sizes[1];            // 4096
  const int V  = in_sizes[0] / Bn;       // 50257

  rankloss_row_kernel<<<Bn, BLOCK, 0, stream>>>(scores, gold, ws, V);
  rankloss_finalize_kernel<<<1, 32, 0, stream>>>(ws, out, Bn);
}